// ReliabLoss_60069412602547
// MI455X (gfx1250) — compile-verified
//
#include <hip/hip_runtime.h>
#include <hip/hip_bf16.h>

// ReliabLoss — CDNA5 (gfx1250) optimized.
//
// Key insight: P = sqrt(Dm)*W*sqrt(Dm) is ELEMENTWISE with diagonal Dm, so P is
// diagonal (P[i,i]=d_i), A is diagonal, and the per-row solve scale cancels in
// U = Fm/Fm.sum(axis=1)  =>  U[i,:] = phi[i,:]/(TAO + s_i),  s_i = sum(labels[i,:]).
// Therefore images / W / d / the 256x256 solve are mathematically dead.
// True working set: preds(256x81)+labels(256x80) ~165KB -> latency bound.
//
// Stage 1: one block per row -> 4 partials/row in d_ws.
// Stage 2: one wave32; 256-wide reductions done with V_WMMA_F32_16X16X4_F32
//          (16x16 matrix x all-ones 4x16, accumulated over K) — full f32 precision.

#define TAO    0.2f
#define EPS    1e-05f
#define NCLASS 80
#define BS     256

typedef __attribute__((ext_vector_type(2))) float v2f;
typedef __attribute__((ext_vector_type(8))) float v8f;

__device__ __forceinline__ float block_sum128(float* red, int tid, float v) {
    red[tid] = v;
    __syncthreads();
    #pragma unroll
    for (int s = 64; s > 0; s >>= 1) {
        if (tid < s) red[tid] += red[tid + s];
        __syncthreads();
    }
    float r = red[0];
    __syncthreads();   // safe reuse of red[]
    return r;
}

// One block per row: compute softmax of preds[row, 0..80], s = sum(labels[row,:]),
// then per-row partial sums of the KL term, emp_rel term, emp_irrel term.
__global__ void __launch_bounds__(128)
row_kernel(const float* __restrict__ preds,
           const float* __restrict__ labels,
           float* __restrict__ part) {
    const int row = blockIdx.x;
    const int tid = threadIdx.x;          // 0..127 (cols 0..80 active)
    __shared__ float red[128];
    __shared__ float sh_p0;

    float e = 0.0f;
    if (tid < NCLASS + 1) e = expf(preds[row * (NCLASS + 1) + tid]);

    float l = 0.0f;
    if (tid >= 1 && tid < NCLASS + 1) l = labels[row * NCLASS + (tid - 1)];

    const float denom = block_sum128(red, tid, e);   // softmax denominator (81 terms)
    const float s_row = block_sum128(red, tid, l);   // #positives in this row

    const float p = e / denom;                       // valid for tid < 81
    if (tid == 0) sh_p0 = p;
    __syncthreads();
    const float p0 = sh_p0;

    const float inv = 1.0f / (TAO + s_row);          // U row normalizer

    float klterm = 0.0f, relterm = 0.0f, irrterm = 0.0f;
    if (tid == 0) {
        const float u0   = TAO * inv;                // always > 0
        const float logp = logf(p + EPS);
        klterm = u0 * logf(u0) - u0 * logp;
    } else if (tid < NCLASS + 1) {
        const float logp = logf(p + EPS);
        if (l == 1.0f) {
            const float u = inv;                     // U>0 entry
            klterm  = u * logf(u) - u * logp;
            relterm = fmaxf(p0 - p, 0.0f);           // relu(py0 - pw)
        } else {                                     // l == 0 -> U=0 -> KL term 0
            const float r = s_row / ((float)NCLASS - s_row);
            irrterm = fmaxf(r * (p - p0), 0.0f);     // relu(r*(pw - py0))
        }
    }

    const float kl_row  = block_sum128(red, tid, klterm);
    const float rel_row = block_sum128(red, tid, relterm);
    const float irr_row = block_sum128(red, tid, irrterm);

    if (tid == 0) {
        part[row * 4 + 0] = kl_row;
        part[row * 4 + 1] = rel_row;
        part[row * 4 + 2] = irr_row;
        part[row * 4 + 3] = s_row;   // positives count for this row
    }
}

// Single wave32. Each of the four 256-long partial arrays is viewed as a 16x16
// f32 matrix M; D = M x ones(16x16) accumulated via 4x V_WMMA_F32_16X16X4_F32
// gives row sums of M replicated in every column. Per-lane sum of the 8 C/D
// VGPRs covers rows 0-7 (lanes 0-15) or 8-15 (lanes 16-31); one shfl_xor(16)
// completes the grand total in full f32.
__global__ void __launch_bounds__(32)
reduce_kernel(const float* __restrict__ part, float* __restrict__ out) {
    const int lane  = threadIdx.x;        // 0..31
    const int m     = lane & 15;          // matrix row handled by this lane (A layout)
    const int khalf = (lane >> 4) * 2;    // lanes 16-31 hold K=+2,+3 of each chunk

    const v2f bones = {1.0f, 1.0f};       // B = all-ones 4x16 (layout-invariant)

    float totals[4];
    #pragma unroll
    for (int q = 0; q < 4; ++q) {
        v8f c = {};
        #pragma unroll
        for (int k0 = 0; k0 < 16; k0 += 4) {
            const int kk = k0 + khalf;
            v2f a;
            a.x = part[(m * 16 + kk)     * 4 + q];
            a.y = part[(m * 16 + kk + 1) * 4 + q];
            // D = A(16x4) x ones(4x16) + C   -> rowsums accumulate
            c = __builtin_amdgcn_wmma_f32_16x16x4_f32(
                    /*neg_a=*/false, a, /*neg_b=*/false, bones,
                    /*c_mod=*/(short)0, c, /*reuse_a=*/false, /*reuse_b=*/false);
        }
        float t = c[0] + c[1] + c[2] + c[3] + c[4] + c[5] + c[6] + c[7];
        t += __shfl_xor(t, 16, 32);       // combine rows 0-7 with rows 8-15
        totals[q] = t;
    }

    if (lane == 0) {
        const float KL    = totals[0];
        const float REL   = totals[1];
        const float IRREL = totals[2];
        const float NPOS  = totals[3];
        const float NNEG  = (float)(BS * NCLASS) - NPOS;

        const float kl       = KL / (float)BS;
        const float emp_rel  = REL / NPOS;
        const float emp_irr  = IRREL / NNEG;
        const float loss_emp = (emp_rel + emp_irr) * (float)BS;

        out[0] = kl;
        out[1] = loss_emp;
    }
}

extern "C" void kernel_launch(void* const* d_in, const int* in_sizes, int n_in,
                              void* d_out, int out_size, void* d_ws, size_t ws_size,
                              hipStream_t stream) {
    // d_in[0] = images (256,2048,7,7) f32 — mathematically dead, never read.
    const float* preds  = (const float*)d_in[1];   // (256, 81)
    const float* labels = (const float*)d_in[2];   // (256, 80)
    float* part = (float*)d_ws;                    // 256*4 floats = 4KB scratch
    float* out  = (float*)d_out;                   // [kl, loss_emp]

    row_kernel<<<BS, 128, 0, stream>>>(preds, labels, part);
    reduce_kernel<<<1, 32, 0, stream>>>(part, out);
}